// Octree_21380347200403
// MI455X (gfx1250) — compile-verified
//
#include <hip/hip_runtime.h>
#include <math.h>

typedef float v2f __attribute__((ext_vector_type(2)));
typedef float v4f __attribute__((ext_vector_type(4)));
typedef float v8f __attribute__((ext_vector_type(8)));

#define RG 128
#define NCH 28
#define NSTEPS 128

// Compute voxel cell + trilinear weights for parameter t along the ray.
__device__ __forceinline__ void cell_of(float t,
                                        float pox, float poy, float poz,
                                        float dvx, float dvy, float dvz,
                                        int& base, float& wx, float& wy, float& wz)
{
    // pos*R - 0.5, clipped to [0, R-1-1e-4] exactly like the reference
    float qx = fminf(fmaxf((pox + t * dvx) * (float)RG - 0.5f, 0.0f), 126.9999f);
    float qy = fminf(fmaxf((poy + t * dvy) * (float)RG - 0.5f, 0.0f), 126.9999f);
    float qz = fminf(fmaxf((poz + t * dvz) * (float)RG - 0.5f, 0.0f), 126.9999f);
    int ix = (int)qx;
    int iy = (int)qy;
    int iz = (int)qz;
    wx = qx - (float)ix;
    wy = qy - (float)iy;
    wz = qz - (float)iz;
    base = ((ix * RG + iy) * RG + iz) * NCH;
}

// Extract element idx (compile-time constant after unrolling) from v4f acc[7].
__device__ __forceinline__ float get28(const v4f* a, int idx)
{
    return a[idx >> 2][idx & 3];
}

__global__ __launch_bounds__(256) void plenoxel_render_kernel(
    const float* __restrict__ rays_o,
    const float* __restrict__ rays_d,
    const float* __restrict__ grid,
    const float* __restrict__ scaling,
    const float* __restrict__ offset,
    float* __restrict__ out,
    int n)
{
    const int ray = blockIdx.x * blockDim.x + threadIdx.x;
    if (ray >= n) return;                 // never divergent for n = 32768 (full waves)
    const int lane = threadIdx.x & 31;    // wave32

    // ---- per-ray setup -------------------------------------------------
    const float ox0 = rays_o[(size_t)ray * 3 + 0];
    const float oy0 = rays_o[(size_t)ray * 3 + 1];
    const float oz0 = rays_o[(size_t)ray * 3 + 2];
    const float rx  = rays_d[(size_t)ray * 3 + 0];
    const float ry  = rays_d[(size_t)ray * 3 + 1];
    const float rz  = rays_d[(size_t)ray * 3 + 2];

    const float sx = scaling[0], sy = scaling[1], sz = scaling[2];
    const float fx = offset[0],  fy = offset[1],  fz = offset[2];

    // unit direction
    const float rn = 1.0f / sqrtf(rx * rx + ry * ry + rz * rz);
    const float ux = rx * rn, uy = ry * rn, uz = rz * rn;

    // scaled origin / direction
    const float pox = ox0 * sx + fx;
    const float poy = oy0 * sy + fy;
    const float poz = oz0 * sz + fz;
    const float dvx = ux * sx, dvy = uy * sy, dvz = uz * sz;

    const float delta_scale = 1.0f / sqrtf(dvx * dvx + dvy * dvy + dvz * dvz);

    // invd = 1 / where(|d| > 1e-9, d, 1e-9)  (sign-replacement matches reference)
    const float ddx = (fabsf(dvx) > 1e-9f) ? dvx : 1e-9f;
    const float ddy = (fabsf(dvy) > 1e-9f) ? dvy : 1e-9f;
    const float ddz = (fabsf(dvz) > 1e-9f) ? dvz : 1e-9f;
    const float ivx = 1.0f / ddx, ivy = 1.0f / ddy, ivz = 1.0f / ddz;

    const float t1x = (0.0f - pox) * ivx, t2x = (1.0f - pox) * ivx;
    const float t1y = (0.0f - poy) * ivy, t2y = (1.0f - poy) * ivy;
    const float t1z = (0.0f - poz) * ivz, t2z = (1.0f - poz) * ivz;

    float tmin = fmaxf(fmaxf(fminf(t1x, t2x), fminf(t1y, t2y)), fminf(t1z, t2z));
    tmin = fmaxf(tmin, 0.0f);
    const float tmax = fminf(fminf(fmaxf(t1x, t2x), fmaxf(t1y, t2y)), fmaxf(t1z, t2z));
    const bool  valid = (tmax > tmin);
    const float dt = valid ? (tmax - tmin) * (1.0f / (float)NSTEPS) : 0.0f;

    // SH basis (degree 2), on unit dirs
    float basis[9];
    {
        const float C0 = 0.28209479177387814f;
        const float C1 = 0.4886025119029199f;
        basis[0] = C0;
        basis[1] = -C1 * uy;
        basis[2] =  C1 * uz;
        basis[3] = -C1 * ux;
        basis[4] =  1.0925484305920792f * ux * uy;
        basis[5] = -1.0925484305920792f * uy * uz;
        basis[6] =  0.31539156525252005f * (2.0f * uz * uz - ux * ux - uy * uy);
        basis[7] = -1.0925484305920792f * ux * uz;
        basis[8] =  0.5462742152960396f * (ux * ux - uy * uy);
    }

    // Masked A operand for the WMMA reduction (built once):
    //   A rows 0-7  = (1,1,0,0)  -> D[0..7][n]  = B[0][n] + B[1][n]  (ray n)
    //   A rows 8-15 = (0,0,1,1)  -> D[8..15][n] = B[2][n] + B[3][n]  (ray n+16)
    // A layout: lane m holds (A[m][0], A[m][1]); lane m+16 holds (A[m][2], A[m][3]).
    //   -> lanes 0-7: (1,1); lanes 8-15: (0,0); lanes 16-23: (0,0); lanes 24-31: (1,1)
    const float aval = ((lane < 8) || (lane >= 24)) ? 1.0f : 0.0f;
    v2f selA; selA.x = aval; selA.y = aval;

    float T = 1.0f;
    float rgb0 = 0.0f, rgb1 = 0.0f, rgb2 = 0.0f;

    // cell for step 0
    int   basec; float wxc, wyc, wzc;
    cell_of(tmin + 0.5f * dt, pox, poy, poz, dvx, dvy, dvz, basec, wxc, wyc, wzc);

    #pragma unroll 1
    for (int i = 0; i < NSTEPS; ++i) {
        // ---- software pipeline: compute next cell, prefetch its corners ----
        int inext = (i + 1 < NSTEPS) ? (i + 1) : i;
        float tn = tmin + ((float)inext + 0.5f) * dt;
        int basen; float wxn, wyn, wzn;
        cell_of(tn, pox, poy, poz, dvx, dvy, dvz, basen, wxn, wyn, wzn);
        {
            const float* pp = grid + (size_t)basen;
            __builtin_prefetch(pp, 0, 3);                                   // (x,y,z..z+1)
            __builtin_prefetch(pp + (size_t)(RG * NCH), 0, 3);              // (x,y+1)
            __builtin_prefetch(pp + (size_t)(RG * RG * NCH), 0, 3);         // (x+1,y)
            __builtin_prefetch(pp + (size_t)((RG * RG + RG) * NCH), 0, 3);  // (x+1,y+1)
        }

        // ---- trilinear gather of 28 channels (8 corners, 7x float4 each) ----
        // Vector accumulators: each corner contributes acc[q] += w4 * c4[q],
        // which lowers to v_pk_fma_f32 pairs with no register repacking.
        v4f acc[7];
        #pragma unroll
        for (int q = 0; q < 7; ++q) acc[q] = (v4f){0.0f, 0.0f, 0.0f, 0.0f};

        const float w1x = wxc, w0x = 1.0f - wxc;
        const float w1y = wyc, w0y = 1.0f - wyc;
        const float w1z = wzc, w0z = 1.0f - wzc;
        const float* gb = grid + (size_t)basec;

        #pragma unroll
        for (int cx = 0; cx < 2; ++cx) {
            const float wxv = cx ? w1x : w0x;
            #pragma unroll
            for (int cy = 0; cy < 2; ++cy) {
                const float wxy = wxv * (cy ? w1y : w0y);
                #pragma unroll
                for (int cz = 0; cz < 2; ++cz) {
                    const float wv = wxy * (cz ? w1z : w0z);
                    const v4f w4 = (v4f){wv, wv, wv, wv};
                    const v4f* c4 = (const v4f*)(gb + (size_t)(((cx * RG + cy) * RG + cz) * NCH));
                    #pragma unroll
                    for (int q = 0; q < 7; ++q) {
                        acc[q] += w4 * c4[q];
                    }
                }
            }
        }

        // ---- SH shading: shuffle-free K-reduction via V_WMMA_F32_16X16X4_F32
        // B is fed with each lane's own products (no cross-lane staging): rows
        // K0/K1 carry lanes 0-15 (rays 0-15), rows K2/K3 carry lanes 16-31
        // (rays 16-31). The masked A selects the matching row pair per output
        // row half, so after chaining 5 chunks:
        //   D[0][n] = sum_j prod_{ray n}[j],  D[8][n] = sum_j prod_{ray n+16}[j]
        // and every lane reads its own ray's sum from D-VGPR0 (acc[0]) directly
        // (lane l<16: M=0,N=l; lane l>=16: M=8,N=l-16). Exact f32 throughout.
        float cols[3];
        #pragma unroll
        for (int ch = 0; ch < 3; ++ch) {
            float prod[10];
            #pragma unroll
            for (int j = 0; j < 9; ++j) prod[j] = basis[j] * get28(acc, ch * 9 + j);
            prod[9] = 0.0f;

            v8f dacc = {0.0f, 0.0f, 0.0f, 0.0f, 0.0f, 0.0f, 0.0f, 0.0f};
            #pragma unroll
            for (int c = 0; c < 5; ++c) {
                v2f B;
                B.x = prod[2 * c + 0];
                B.y = prod[2 * c + 1];
                dacc = __builtin_amdgcn_wmma_f32_16x16x4_f32(
                    false, selA, false, B, (short)0, dacc, false, false);
            }
            const float s = dacc[0];
            cols[ch] = 1.0f / (1.0f + expf(-s));     // sigmoid
        }

        // ---- compositing ---------------------------------------------------
        const float sigma = fmaxf(acc[6][3], 0.0f);  // channel 27; dt==0 kills invalid rays
        const float alpha = 1.0f - expf(-sigma * dt * delta_scale);
        const float wA = T * alpha;
        rgb0 += wA * cols[0];
        rgb1 += wA * cols[1];
        rgb2 += wA * cols[2];
        T *= (1.0f - alpha);

        basec = basen; wxc = wxn; wyc = wyn; wzc = wzn;
    }

    // rgb + T * BG  (BG = 1.0)
    out[(size_t)ray * 3 + 0] = rgb0 + T;
    out[(size_t)ray * 3 + 1] = rgb1 + T;
    out[(size_t)ray * 3 + 2] = rgb2 + T;
}

extern "C" void kernel_launch(void* const* d_in, const int* in_sizes, int n_in,
                              void* d_out, int out_size, void* d_ws, size_t ws_size,
                              hipStream_t stream) {
    const float* rays_o  = (const float*)d_in[0];
    const float* rays_d  = (const float*)d_in[1];
    const float* grid    = (const float*)d_in[2];
    const float* scaling = (const float*)d_in[3];
    const float* offset  = (const float*)d_in[4];
    float* out = (float*)d_out;

    const int n = in_sizes[0] / 3;                 // 32768 rays
    const int block = 256;                         // 8 wave32 waves per block
    const int ngrid = (n + block - 1) / block;     // 128 blocks

    plenoxel_render_kernel<<<dim3(ngrid), dim3(block), 0, stream>>>(
        rays_o, rays_d, grid, scaling, offset, out, n);
}